// rnn_80702435492167
// MI455X (gfx1250) — compile-verified
//
#include <hip/hip_runtime.h>
#include <math.h>

#define NN   8192
#define NZ   256
#define NC   128
#define DTC  0.1f

// output layout: [x_new(NN), r_new(NN), z_new(NZ), c_new(NC), eps_new(NZ)]
#define OUT_X 0
#define OUT_R NN
#define OUT_Z (2 * NN)
#define OUT_C (2 * NN + NZ)
#define OUT_E (2 * NN + NZ + NC)

#define KSPLIT 8
#define KCHUNK (NN / KSPLIT)   // 1024

typedef __attribute__((ext_vector_type(2))) float v2f;
typedef __attribute__((ext_vector_type(4))) float v4f;
typedef __attribute__((ext_vector_type(8))) float v8f;

// One wave computes y[row0..row0+15] += W[rows, 0..K) @ vec[0..K)
// using V_WMMA_F32_16X16X4_F32.  W is row-major with leading dim `ld`.
// Result: lane 0 holds y[row0+0..7] in acc[0..7], lane 16 holds y[row0+8..15].
// NT=true marks the weight stream non-temporal (single-use, > L2 capacity).
template <bool NT>
__device__ __forceinline__ v8f wmma_gemv(v8f acc,
                                         const float* __restrict__ W, int ld,
                                         int row0,
                                         const float* __restrict__ vec, int K) {
  const int lane = threadIdx.x & 31;
  const int m    = lane & 15;   // row within 16-row tile
  const int hl   = lane >> 4;   // K half-select (ISA A layout: lanes>=16 carry K2,K3)
  const float* wp = W + (size_t)(row0 + m) * (size_t)ld + 4 * hl;
  const float* vp = vec + 4 * hl;
#pragma unroll 8
  for (int k = 0; k < K; k += 8) {
    const v4f w = NT ? __builtin_nontemporal_load((const v4f*)(wp + k))
                     : *(const v4f*)(wp + k);     // 16B/lane, 512B/wave of W
    const v4f bv = *(const v4f*)(vp + k);         // broadcast vector from LDS
    v2f a0 = {w.x, w.y}, b0 = {bv.x, bv.y};
    v2f a1 = {w.z, w.w}, b1 = {bv.z, bv.w};
    acc = __builtin_amdgcn_wmma_f32_16x16x4_f32(false, a0, false, b0,
                                                (short)0, acc, false, false);
    acc = __builtin_amdgcn_wmma_f32_16x16x4_f32(false, a1, false, b1,
                                                (short)0, acc, false, false);
  }
  return acc;
}

// ---- stage 1a: z partials:  zpart[ks][row] = W_rz[row, ks*1024 .. ) @ r_chunk
__global__ void k_zpart(const float* __restrict__ W_rz,
                        const float* __restrict__ x,
                        const float* __restrict__ b,
                        float* __restrict__ zpart) {
  __shared__ __align__(16) float s_r[KCHUNK];
  const int tid = threadIdx.x;
  const int k0  = blockIdx.y * KCHUNK;
  for (int j = tid; j < KCHUNK; j += 256)
    s_r[j] = tanhf(x[k0 + j] + b[k0 + j]);        // r = tanh(x+b), chunk
  __syncthreads();
  const int row0 = blockIdx.x * 128 + (tid >> 5) * 16;
  v8f acc = {0.f, 0.f, 0.f, 0.f, 0.f, 0.f, 0.f, 0.f};
  acc = wmma_gemv<false>(acc, W_rz + k0, NN, row0, s_r, KCHUNK);
  const int lane = tid & 31;
  if ((lane & 15) == 0) {
    const int rbase = row0 + (lane >> 4) * 8;
    float* dst = zpart + blockIdx.y * NZ;
#pragma unroll
    for (int v = 0; v < 8; ++v) dst[rbase + v] = acc[v];
  }
}

// ---- stage 1b: W_rr partials (the 256 MB stream): 64 row-groups x 8 K-splits
__global__ void k_rrpart(const float* __restrict__ W_rr,
                         const float* __restrict__ x,
                         const float* __restrict__ b,
                         float* __restrict__ rrpart) {
  __shared__ __align__(16) float s_r[KCHUNK];
  const int tid = threadIdx.x;
  const int k0  = blockIdx.y * KCHUNK;
  for (int j = tid; j < KCHUNK; j += 256)
    s_r[j] = tanhf(x[k0 + j] + b[k0 + j]);
  __syncthreads();
  const int row0 = blockIdx.x * 128 + (tid >> 5) * 16;
  v8f acc = {0.f, 0.f, 0.f, 0.f, 0.f, 0.f, 0.f, 0.f};
  acc = wmma_gemv<true>(acc, W_rr + k0, NN, row0, s_r, KCHUNK);  // NT stream
  const int lane = tid & 31;
  if ((lane & 15) == 0) {
    const int rbase = row0 + (lane >> 4) * 8;
    float* dst = rrpart + blockIdx.y * NN;
#pragma unroll
    for (int v = 0; v < 8; ++v) dst[rbase + v] = acc[v];
  }
}

// ---- stage 2: combine partials + small GEMVs + Euler update + tanh
__global__ void k_final(const float* __restrict__ x,
                        const float* __restrict__ b,
                        const float* __restrict__ eps,
                        const float* __restrict__ cvec,
                        const float* __restrict__ W_epsr,
                        const float* __restrict__ W_zr,
                        const float* __restrict__ W_cr,
                        const float* __restrict__ zpart,
                        const float* __restrict__ rrpart,
                        float* __restrict__ out) {
  __shared__ __align__(16) float s_eps[NZ];
  __shared__ __align__(16) float s_z[NZ];
  __shared__ __align__(16) float s_c[NC];
  const int tid = threadIdx.x;
  {
    float zs = 0.f;                      // fixed-order reduction -> deterministic
#pragma unroll
    for (int s = 0; s < KSPLIT; ++s) zs += zpart[s * NZ + tid];
    s_z[tid]   = zs;
    s_eps[tid] = eps[tid];
    if (tid < NC) s_c[tid] = cvec[tid];
  }
  __syncthreads();
  const int row0 = blockIdx.x * 128 + (tid >> 5) * 16;
  v8f acc = {0.f, 0.f, 0.f, 0.f, 0.f, 0.f, 0.f, 0.f};
  acc = wmma_gemv<false>(acc, W_epsr, NZ, row0, s_eps, NZ);
  acc = wmma_gemv<false>(acc, W_zr,   NZ, row0, s_z,   NZ);
  acc = wmma_gemv<false>(acc, W_cr,   NC, row0, s_c,   NC);
  const int lane = tid & 31;
  if ((lane & 15) == 0) {
    const int rbase = row0 + (lane >> 4) * 8;
#pragma unroll
    for (int v = 0; v < 8; ++v) {
      const int i = rbase + v;
      float tot = acc[v];
#pragma unroll
      for (int s = 0; s < KSPLIT; ++s) tot += rrpart[s * NN + i];
      const float xv = x[i];
      const float xn = xv + DTC * (tot - xv);   // x + dt*(-x + Wr+...)/tau
      out[OUT_X + i] = xn;
      out[OUT_R + i] = tanhf(xn + b[i]);
    }
  }
}

// ---- stage 3: z_new / c_new partials from r_new (read back from out[NN..2NN))
__global__ void k_outpart(const float* __restrict__ W_rz,
                          const float* __restrict__ W_rc,
                          const float* __restrict__ r_new,
                          float* __restrict__ znpart,
                          float* __restrict__ cnpart) {
  __shared__ __align__(16) float s_r[KCHUNK];
  const int tid = threadIdx.x;
  const int k0  = blockIdx.y * KCHUNK;
  for (int j = tid; j < KCHUNK; j += 256) s_r[j] = r_new[k0 + j];
  __syncthreads();
  const int wave = tid >> 5;
  const int lane = tid & 31;
  v8f acc = {0.f, 0.f, 0.f, 0.f, 0.f, 0.f, 0.f, 0.f};
  if (blockIdx.x < 2) {                       // z_new rows (2 x 128)
    const int row0 = blockIdx.x * 128 + wave * 16;
    acc = wmma_gemv<false>(acc, W_rz + k0, NN, row0, s_r, KCHUNK);
    if ((lane & 15) == 0) {
      const int rbase = row0 + (lane >> 4) * 8;
#pragma unroll
      for (int v = 0; v < 8; ++v) znpart[blockIdx.y * NZ + rbase + v] = acc[v];
    }
  } else {                                    // c_new rows (128)
    const int row0 = wave * 16;
    acc = wmma_gemv<false>(acc, W_rc + k0, NN, row0, s_r, KCHUNK);
    if ((lane & 15) == 0) {
      const int rbase = row0 + (lane >> 4) * 8;
#pragma unroll
      for (int v = 0; v < 8; ++v) cnpart[blockIdx.y * NC + rbase + v] = acc[v];
    }
  }
}

// ---- stage 4: finalize z_new, c_new, eps_new
__global__ void k_epilog(const float* __restrict__ znpart,
                         const float* __restrict__ cnpart,
                         const float* __restrict__ z_tilde,
                         float* __restrict__ out) {
  const int tid = threadIdx.x;   // 256 threads
  float zn = 0.f;
#pragma unroll
  for (int s = 0; s < KSPLIT; ++s) zn += znpart[s * NZ + tid];
  out[OUT_Z + tid] = zn;
  out[OUT_E + tid] = zn - z_tilde[tid];
  if (tid < NC) {
    float cn = 0.f;
#pragma unroll
    for (int s = 0; s < KSPLIT; ++s) cn += cnpart[s * NC + tid];
    out[OUT_C + tid] = cn;
  }
}

extern "C" void kernel_launch(void* const* d_in, const int* in_sizes, int n_in,
                              void* d_out, int out_size, void* d_ws, size_t ws_size,
                              hipStream_t stream) {
  (void)in_sizes; (void)n_in; (void)out_size; (void)ws_size;
  const float* x       = (const float*)d_in[0];
  const float* eps     = (const float*)d_in[1];
  const float* c       = (const float*)d_in[2];
  const float* z_tilde = (const float*)d_in[3];
  const float* W_rr    = (const float*)d_in[4];
  const float* W_zr    = (const float*)d_in[5];
  const float* W_cr    = (const float*)d_in[6];
  const float* W_epsr  = (const float*)d_in[7];
  const float* W_rz    = (const float*)d_in[8];
  const float* W_rc    = (const float*)d_in[9];
  const float* b       = (const float*)d_in[10];
  float* out = (float*)d_out;

  float* ws     = (float*)d_ws;
  float* zpart  = ws;                        // KSPLIT*NZ = 2048 floats
  float* rrpart = zpart  + KSPLIT * NZ;      // KSPLIT*NN = 65536 floats
  float* znpart = rrpart + KSPLIT * NN;      // KSPLIT*NZ
  float* cnpart = znpart + KSPLIT * NZ;      // KSPLIT*NC

  k_zpart  <<<dim3(2, KSPLIT),  256, 0, stream>>>(W_rz, x, b, zpart);
  k_rrpart <<<dim3(64, KSPLIT), 256, 0, stream>>>(W_rr, x, b, rrpart);
  k_final  <<<64, 256, 0, stream>>>(x, b, eps, c, W_epsr, W_zr, W_cr,
                                    zpart, rrpart, out);
  k_outpart<<<dim3(3, KSPLIT),  256, 0, stream>>>(W_rz, W_rc, out + NN,
                                                  znpart, cnpart);
  k_epilog <<<1, 256, 0, stream>>>(znpart, cnpart, z_tilde, out);
}